// SparseDimAttention_77988016161046
// MI455X (gfx1250) — compile-verified
//
#include <hip/hip_runtime.h>
#include <math.h>

#define B_SZ 512
#define L_SZ 33
#define D_SZ 4096
#define P_SZ 64
#define K_SZ 512
#define H_SZ 128
#define LN_EPS 1e-5f

typedef float v2f __attribute__((ext_vector_type(2)));
typedef float v8f __attribute__((ext_vector_type(8)));

__device__ __forceinline__ unsigned f2key(float f) {
    unsigned u = __float_as_uint(f);
    return (u & 0x80000000u) ? ~u : (u | 0x80000000u);
}
__device__ __forceinline__ float key2f(unsigned k) {
    unsigned u = (k & 0x80000000u) ? (k ^ 0x80000000u) : ~k;
    return __uint_as_float(u);
}

// Kernel 1: per-batch scores -> radix top-k -> softmax-weighted xbar ->
// Wp@xbar+bp -> LayerNorm -> z (B x 64) into workspace.
__global__ __launch_bounds__(256) void sda_select_kernel(
    const float* __restrict__ x, const float* __restrict__ Wp,
    const float* __restrict__ bp, const float* __restrict__ ws,
    const float* __restrict__ bs, const float* __restrict__ gamma,
    const float* __restrict__ beta, float* __restrict__ z_out)
{
    __shared__ float sV[L_SZ];
    __shared__ float sC;
    __shared__ float sS[D_SZ];           // 16 KB scores
    __shared__ unsigned sHist[256];
    __shared__ unsigned sMaxKey;
    __shared__ unsigned sPrefix;
    __shared__ int sRemK;
    __shared__ int sSelCnt, sEqCnt;
    __shared__ int sIdx[K_SZ];
    __shared__ float sE[K_SZ];
    __shared__ float sSum;
    __shared__ float sXbar[L_SZ];
    __shared__ float sOut[P_SZ];
    __shared__ float sMu, sRstd;

    const int tid = threadIdx.x;
    const int b = blockIdx.x;
    const float* __restrict__ xb = x + (size_t)b * L_SZ * D_SZ;

    // v[l] = sum_p ws[p]*Wp[p][l];  c = ws.bp + bs
    if (tid < L_SZ) {
        float acc = 0.f;
        for (int p = 0; p < P_SZ; ++p) acc += ws[p] * Wp[p * L_SZ + tid];
        sV[tid] = acc;
    }
    if (tid == L_SZ) {
        float acc = bs[0];
        for (int p = 0; p < P_SZ; ++p) acc += ws[p] * bp[p];
        sC = acc;
    }
    if (tid == 0) { sMaxKey = 0u; sSum = 0.f; sSelCnt = 0; sEqCnt = 0; }
    if (tid < L_SZ) sXbar[tid] = 0.f;
    __syncthreads();

    // ---- scores: one streaming pass over x[b], float4-vectorized ----
    float vreg[L_SZ];
    #pragma unroll
    for (int l = 0; l < L_SZ; ++l) vreg[l] = sV[l];
    const float cc = sC;
    unsigned localMax = 0u;
    #pragma unroll
    for (int i = 0; i < D_SZ / (256 * 4); ++i) {   // 4 iterations
        int d4 = (tid + i * 256) * 4;
        float a0 = cc, a1 = cc, a2 = cc, a3 = cc;
        #pragma unroll
        for (int l = 0; l < L_SZ; ++l) {
            const float4 xv = *(const float4*)(xb + (size_t)l * D_SZ + d4);
            float vl = vreg[l];
            a0 += vl * xv.x;
            a1 += vl * xv.y;
            a2 += vl * xv.z;
            a3 += vl * xv.w;
        }
        float4 sv = make_float4(a0, a1, a2, a3);
        *(float4*)(sS + d4) = sv;
        unsigned k0 = f2key(a0), k1 = f2key(a1), k2 = f2key(a2), k3 = f2key(a3);
        unsigned kmax = k0 > k1 ? k0 : k1;
        if (k2 > kmax) kmax = k2;
        if (k3 > kmax) kmax = k3;
        if (kmax > localMax) localMax = kmax;
    }
    atomicMax(&sMaxKey, localMax);
    __syncthreads();
    const float m = key2f(sMaxKey);

    // ---- radix select: key of the K-th largest score ----
    unsigned prefix = 0u;
    int remK = K_SZ;
    for (int shift = 24; shift >= 0; shift -= 8) {
        sHist[tid] = 0u;                 // 256 threads == 256 bins
        __syncthreads();
        unsigned hiMask = (shift == 24) ? 0u : (0xFFFFFFFFu << (shift + 8));
        for (int i = 0; i < D_SZ / 256; ++i) {
            unsigned k = f2key(sS[tid + i * 256]);
            if ((k & hiMask) == (prefix & hiMask))
                atomicAdd(&sHist[(k >> shift) & 255u], 1u);
        }
        __syncthreads();
        if (tid == 0) {
            int cum = 0;
            for (int g = 255; g >= 0; --g) {
                int c = (int)sHist[g];
                if (cum + c >= remK) {
                    sPrefix = prefix | ((unsigned)g << shift);
                    sRemK = remK - cum;
                    break;
                }
                cum += c;
            }
        }
        __syncthreads();
        prefix = sPrefix;
        remK = sRemK;
        __syncthreads();
    }
    const unsigned T = prefix;           // threshold key
    const int nEq = remK;                // #elements == T to take

    // ---- selection + softmax numerators ----
    for (int i = 0; i < D_SZ / 256; ++i) {
        int d = tid + i * 256;
        float s = sS[d];
        unsigned k = f2key(s);
        int pos = -1;
        if (k > T) pos = atomicAdd(&sSelCnt, 1);
        else if (k == T) {
            int q = atomicAdd(&sEqCnt, 1);
            if (q < nEq) pos = atomicAdd(&sSelCnt, 1);
        }
        if (pos >= 0) {
            float e = expf(s - m);
            sIdx[pos] = d;
            sE[pos] = e;
            atomicAdd(&sSum, e);
        }
    }
    __syncthreads();
    const float invS = 1.0f / sSum;

    // ---- xbar[l] = sum_k e_k * x[b,l,idx_k]  (gather; L2-resident) ----
    float priv[L_SZ];
    #pragma unroll
    for (int l = 0; l < L_SZ; ++l) priv[l] = 0.f;
    for (int k = tid; k < K_SZ; k += 256) {
        int d = sIdx[k];
        float e = sE[k];
        #pragma unroll
        for (int l = 0; l < L_SZ; ++l)
            priv[l] += e * xb[(size_t)l * D_SZ + d];
    }
    #pragma unroll
    for (int l = 0; l < L_SZ; ++l) {
        float v = priv[l];
        v += __shfl_xor(v, 16, 32);
        v += __shfl_xor(v, 8, 32);
        v += __shfl_xor(v, 4, 32);
        v += __shfl_xor(v, 2, 32);
        v += __shfl_xor(v, 1, 32);
        if ((tid & 31) == 0) atomicAdd(&sXbar[l], v);
    }
    __syncthreads();

    // ---- out[p] = bp[p] + invS * Wp[p,:].xbar ; then LayerNorm ----
    if (tid < P_SZ) {
        float acc = 0.f;
        #pragma unroll
        for (int l = 0; l < L_SZ; ++l) acc += Wp[tid * L_SZ + l] * sXbar[l];
        sOut[tid] = bp[tid] + invS * acc;
    }
    __syncthreads();
    if (tid == 0) {
        float mu = 0.f;
        for (int p = 0; p < P_SZ; ++p) mu += sOut[p];
        mu *= (1.0f / P_SZ);
        float var = 0.f;
        for (int p = 0; p < P_SZ; ++p) { float dd = sOut[p] - mu; var += dd * dd; }
        var *= (1.0f / P_SZ);
        sMu = mu;
        sRstd = rsqrtf(var + LN_EPS);
    }
    __syncthreads();
    if (tid < P_SZ) {
        float z = (sOut[tid] - sMu) * sRstd * gamma[tid] + beta[tid];
        z_out[(size_t)b * P_SZ + tid] = z;
    }
}

// Kernel 2: head GEMM z(512x64) @ W1^T(64x128) with V_WMMA_F32_16X16X4_F32,
// fused bias + exact GELU, then 128->2 projection through LDS.
__global__ __launch_bounds__(256) void sda_head_kernel(
    const float* __restrict__ z, const float* __restrict__ W1,
    const float* __restrict__ b1, const float* __restrict__ W2,
    const float* __restrict__ b2, float* __restrict__ out)
{
    __shared__ float sG[16 * H_SZ];      // 8 KB GELU activations
    const int tid = threadIdx.x;
    const int wave = tid >> 5;           // 8 waves: N tiles
    const int lane = tid & 31;
    const int m0 = blockIdx.x * 16;      // 32 blocks: M tiles
    const int n0 = wave * 16;
    const int half = lane >> 4;          // 0: lanes 0-15, 1: lanes 16-31
    const int l15 = lane & 15;
    const int koff = half * 2;           // ISA 16x4 f32 frag: high lanes hold K+2

    const float* __restrict__ zrow = z + (size_t)(m0 + l15) * P_SZ;
    const float* __restrict__ wrow = W1 + (size_t)(n0 + l15) * P_SZ; // B[k][n]=W1[n*P+k]

    v8f acc = {};
    #pragma unroll
    for (int kk = 0; kk < P_SZ; kk += 4) {
        v2f a, bf;
        a.x = zrow[kk + koff];
        a.y = zrow[kk + koff + 1];
        bf.x = wrow[kk + koff];
        bf.y = wrow[kk + koff + 1];
        acc = __builtin_amdgcn_wmma_f32_16x16x4_f32(
            false, a, false, bf, (short)0, acc, false, false);
    }

    float bias = b1[n0 + l15];
    #pragma unroll
    for (int r = 0; r < 8; ++r) {
        int mrow = r + half * 8;         // D layout: vgpr r -> M=r (low) / M=r+8 (high)
        float g = acc[r] + bias;
        g = 0.5f * g * (1.0f + erff(g * 0.70710678118654752f)); // exact GELU
        sG[mrow * H_SZ + n0 + l15] = g;
    }
    __syncthreads();

    if (tid < 32) {
        int mrow = tid & 15;
        int j = tid >> 4;
        float s = b2[j];
        const float* grow = sG + mrow * H_SZ;
        const float* w2r = W2 + j * H_SZ;
        #pragma unroll 8
        for (int h = 0; h < H_SZ; ++h) s += grow[h] * w2r[h];
        out[(size_t)(m0 + mrow) * 2 + j] = s;
    }
}

extern "C" void kernel_launch(void* const* d_in, const int* in_sizes, int n_in,
                              void* d_out, int out_size, void* d_ws, size_t ws_size,
                              hipStream_t stream) {
    const float* x     = (const float*)d_in[0];
    const float* Wp    = (const float*)d_in[1];
    const float* bp    = (const float*)d_in[2];
    const float* ws    = (const float*)d_in[3];
    const float* bs    = (const float*)d_in[4];
    const float* gamma = (const float*)d_in[5];
    const float* beta  = (const float*)d_in[6];
    const float* W1    = (const float*)d_in[7];
    const float* b1    = (const float*)d_in[8];
    const float* W2    = (const float*)d_in[9];
    const float* b2    = (const float*)d_in[10];

    float* z   = (float*)d_ws;           // 512*64*4 = 128 KB scratch
    float* out = (float*)d_out;

    sda_select_kernel<<<B_SZ, 256, 0, stream>>>(x, Wp, bp, ws, bs, gamma, beta, z);
    sda_head_kernel<<<B_SZ / 16, 256, 0, stream>>>(z, W1, b1, W2, b2, out);
}